// ATMOp_661424963816
// MI455X (gfx1250) — compile-verified
//
#include <hip/hip_runtime.h>

typedef __attribute__((ext_vector_type(2))) float v2f;
typedef __attribute__((ext_vector_type(8))) float v8f;

#define Bn 8
#define Cn 64
#define On 64
#define Hn 256
#define Wn 256

// Branchless 1-D bilinear sample along W with zero padding outside [0, Wn-1].
// cb = flat element offset of (b, c, h, 0) in a [B,C,H,W] tensor.
__device__ __forceinline__ float sample_w(const float* __restrict__ inp,
                                          const float* __restrict__ off,
                                          int cb, int w)
{
    float x  = off[cb + w] + (float)w;   // sample coordinate
    float xf = floorf(x);
    float lx = x - xf;
    int x0 = (int)xf;
    int x1 = x0 + 1;
    int x0c = min(max(x0, 0), Wn - 1);
    int x1c = min(max(x1, 0), Wn - 1);
    float v0 = inp[cb + x0c];            // always in-bounds load
    float v1 = inp[cb + x1c];
    v0 = (x0 >= 0 && x0 < Wn) ? v0 : 0.0f;   // v_cndmask, no divergence
    v1 = (x1 >= 0 && x1 < Wn) ? v1 : 0.0f;
    return v0 * (1.0f - lx) + v1 * lx;
}

// One wave -> 16 pixels x all 64 outputs via 4 WMMA M-tiles.
// A (16x4) = weight tile [outputs x K], B (4x16) = sampled tile [K x pixels].
__global__ __launch_bounds__(256) void deform1x1_wmma(
    const float* __restrict__ inp,
    const float* __restrict__ off,
    const float* __restrict__ wgt,   // [O, C] row-major
    const float* __restrict__ bias,  // [O]
    float* __restrict__ out)         // [B, O, H, W]
{
    const int lane = threadIdx.x & 31;
    const int wave = threadIdx.x >> 5;
    const int m    = lane & 15;   // A: output row in tile / B: pixel in tile
    const int kh   = lane >> 4;   // which K-pair of the K=4 step this lane holds

    // 16 consecutive pixels along W for this wave (16 | Wn, never crosses rows)
    const int pixBase = (blockIdx.x * 8 + wave) * 16;
    const int b  = pixBase >> 16;        // / (Hn*Wn)
    const int hw = pixBase & 0xFFFF;
    const int h  = hw >> 8;              // / Wn
    const int w0 = hw & 0xFF;
    const int w  = w0 + m;

    v8f acc0 = {}; v8f acc1 = {}; v8f acc2 = {}; v8f acc3 = {};

#pragma unroll
    for (int s = 0; s < 16; ++s) {
        const int k  = 4 * s + 2 * kh;               // this lane's first K
        const int cb = ((b * Cn + k) * Hn + h) * Wn; // (b, k, h, 0)

        // B fragment: sampled[k][pixel], sampled[k+1][pixel]
        v2f bf;
        bf.x = sample_w(inp, off, cb,           w);
        bf.y = sample_w(inp, off, cb + Hn * Wn, w);

        // A fragments for the 4 output tiles: weight[o][k], weight[o][k+1]
        v2f a0 = *(const v2f*)(wgt + ( 0 + m) * Cn + k);
        v2f a1 = *(const v2f*)(wgt + (16 + m) * Cn + k);
        v2f a2 = *(const v2f*)(wgt + (32 + m) * Cn + k);
        v2f a3 = *(const v2f*)(wgt + (48 + m) * Cn + k);

        acc0 = __builtin_amdgcn_wmma_f32_16x16x4_f32(false, a0, false, bf, (short)0, acc0, false, false);
        acc1 = __builtin_amdgcn_wmma_f32_16x16x4_f32(false, a1, false, bf, (short)0, acc1, false, false);
        acc2 = __builtin_amdgcn_wmma_f32_16x16x4_f32(false, a2, false, bf, (short)0, acc2, false, false);
        acc3 = __builtin_amdgcn_wmma_f32_16x16x4_f32(false, a3, false, bf, (short)0, acc3, false, false);
    }

    // D layout: VGPR r -> (M = r + 8*kh, N = m). M = output-in-tile, N = pixel.
    // Lanes 0-15 and 16-31 each store 64B contiguous runs along W: coalesced.
#pragma unroll
    for (int t = 0; t < 4; ++t) {
        v8f acc = (t == 0) ? acc0 : (t == 1) ? acc1 : (t == 2) ? acc2 : acc3;
#pragma unroll
        for (int r = 0; r < 8; ++r) {
            const int o = t * 16 + r + 8 * kh;
            out[((b * On + o) * Hn + h) * Wn + w0 + m] = acc[r] + bias[o];
        }
    }
}

extern "C" void kernel_launch(void* const* d_in, const int* in_sizes, int n_in,
                              void* d_out, int out_size, void* d_ws, size_t ws_size,
                              hipStream_t stream) {
    const float* inp  = (const float*)d_in[0];
    const float* off  = (const float*)d_in[1];
    const float* wgt  = (const float*)d_in[2];
    const float* bias = (const float*)d_in[3];
    float* out = (float*)d_out;

    // P = B*H*W = 524288 pixels; 16 pixels/wave, 8 waves/block -> 4096 blocks
    const int blocks = (Bn * Hn * Wn) / 128;
    deform1x1_wmma<<<blocks, 256, 0, stream>>>(inp, off, wgt, bias, out);
}